// Convolution_4200478015529
// MI455X (gfx1250) — compile-verified
//
#include <hip/hip_runtime.h>
#include <hip/hip_bf16.h>
#include <stdint.h>

typedef __attribute__((ext_vector_type(2))) float v2f;
typedef __attribute__((ext_vector_type(8))) float v8f;

#define GCN_NODES 50000
#define GCN_INF   128
#define GCN_OUTF  128

// ---------------------------------------------------------------------------
// Kernel 1: zero the output accumulator (harness poisons d_out with 0xAA).
// ---------------------------------------------------------------------------
__global__ __launch_bounds__(256) void gcn_zero_f4(float4* __restrict__ p, int n4) {
    int i = blockIdx.x * blockDim.x + threadIdx.x;
    if (i < n4) p[i] = make_float4(0.f, 0.f, 0.f, 0.f);
}

// ---------------------------------------------------------------------------
// Kernel 2: weighted = input[50000x128] @ weight[128x128], fp32 WMMA.
//
// One wave computes one 16x16 output tile using V_WMMA_F32_16X16X4_F32,
// K=128 -> 32 WMMA ops. Block = 256 threads = 8 waves; wave w owns N-tile w,
// block b owns M-tile b (50000 = 3125 * 16 exactly, 128 = 8 * 16 exactly).
//
// fp32 A-matrix 16x4 layout (ISA 7.12.2): lane = hi*16+lo, row M = lo,
//   VGPR v holds K = v + 2*hi  -> per-lane float2 load at A[m][k0 + 2*hi].
// fp32 B-matrix 4x16: col N = lo, VGPR v holds K = v + 2*hi.
// C/D 16x16: VGPR v holds row M = v + 8*hi, col N = lo.
// ---------------------------------------------------------------------------
__global__ __launch_bounds__(256) void gcn_gemm_wmma_f32(
        const float* __restrict__ A,   // [50000,128] row-major
        const float* __restrict__ B,   // [128,128]   row-major
        float* __restrict__ W)         // [50000,128] row-major
{
    const int wave  = threadIdx.x >> 5;   // 0..7  -> N tile
    const int lane  = threadIdx.x & 31;
    const int hi    = lane >> 4;          // 0 or 1
    const int lo    = lane & 15;
    const int mtile = blockIdx.x;         // 0..3124

    const int m = mtile * 16 + lo;        // A row for this lane
    const int n = wave * 16 + lo;         // B/D col for this lane

    const float* arow = A + (size_t)m * GCN_INF + 2 * hi;          // + k0
    const float* bcol = B + (size_t)(2 * hi) * GCN_OUTF + n;       // + k0*128

    v8f acc = {};
#pragma unroll
    for (int k0 = 0; k0 < GCN_INF; k0 += 4) {
        // A fragment: VGPR0 = A[m][k0+2*hi], VGPR1 = A[m][k0+2*hi+1]
        v2f a = *(const v2f*)(arow + k0);
        // B fragment: VGPR0 = B[k0+2*hi][n], VGPR1 = B[k0+2*hi+1][n]
        v2f b;
        b.x = bcol[(size_t)k0 * GCN_OUTF];
        b.y = bcol[(size_t)(k0 + 1) * GCN_OUTF];
        // 8 args: (neg_a, A, neg_b, B, c_mod, C, reuse_a, reuse_b)
        acc = __builtin_amdgcn_wmma_f32_16x16x4_f32(
                false, a, false, b, (short)0, acc, false, false);
    }

    // Store D: VGPR v -> row (mtile*16 + v + 8*hi), col (wave*16 + lo)
    float* wout = W + (size_t)(mtile * 16 + 8 * hi) * GCN_OUTF + n;
#pragma unroll
    for (int v = 0; v < 8; ++v) {
        wout[(size_t)v * GCN_OUTF] = acc[v];
    }
}

// ---------------------------------------------------------------------------
// Kernel 3: out[r] += val_e * weighted[c]   (edge-parallel scatter)
//
// One wave per edge: lane loads float4 (features lane*4..lane*4+3) -> the
// wave's gather is one coalesced 512B line from weighted[c] (L2-resident:
// weighted is 25.6MB inside the 192MB L2), then 4 f32 atomics per lane.
// ---------------------------------------------------------------------------
__global__ __launch_bounds__(256) void gcn_spmm_scatter(
        const float*   __restrict__ weighted,  // [50000,128]
        const int64_t* __restrict__ rows,      // [E]
        const int64_t* __restrict__ cols,      // [E]
        const float*   __restrict__ vals,      // [E]
        float*         __restrict__ out,       // [50000,128]
        int nE)
{
    const int wavesPerBlock = blockDim.x >> 5;
    const int e    = blockIdx.x * wavesPerBlock + (threadIdx.x >> 5);
    const int lane = threadIdx.x & 31;
    if (e >= nE) return;

    const int   r = (int)rows[e];
    const int   c = (int)cols[e];
    const float v = vals[e];

    const float4 w = *((const float4*)(weighted + (size_t)c * GCN_OUTF) + lane);
    float* dst = out + (size_t)r * GCN_OUTF + lane * 4;

    atomicAdd(dst + 0, v * w.x);
    atomicAdd(dst + 1, v * w.y);
    atomicAdd(dst + 2, v * w.z);
    atomicAdd(dst + 3, v * w.w);
}

// ---------------------------------------------------------------------------
// Launch: zero(out) -> gemm(ws) -> scatter(out). Same stream => ordered.
// ---------------------------------------------------------------------------
extern "C" void kernel_launch(void* const* d_in, const int* in_sizes, int n_in,
                              void* d_out, int out_size, void* d_ws, size_t ws_size,
                              hipStream_t stream) {
    const float*   input  = (const float*)  d_in[0];   // [50000,128] f32
    const float*   weight = (const float*)  d_in[1];   // [128,128]   f32
    const int64_t* rows   = (const int64_t*)d_in[2];   // [E] i64
    const int64_t* cols   = (const int64_t*)d_in[3];   // [E] i64
    const float*   vals   = (const float*)  d_in[4];   // [E] f32

    float* out      = (float*)d_out;
    float* weighted = (float*)d_ws;                    // 25.6 MB scratch

    // 1) zero output
    const int n4 = GCN_NODES * GCN_OUTF / 4;
    gcn_zero_f4<<<(n4 + 255) / 256, 256, 0, stream>>>((float4*)out, n4);

    // 2) dense transform via fp32 WMMA
    gcn_gemm_wmma_f32<<<GCN_NODES / 16, 256, 0, stream>>>(input, weight, weighted);

    // 3) edge scatter with atomics
    const int nE = in_sizes[2];
    const int wavesPerBlock = 8;  // 256 threads
    const int blocks = (nE + wavesPerBlock - 1) / wavesPerBlock;
    gcn_spmm_scatter<<<blocks, 256, 0, stream>>>(weighted, rows, cols, vals, out, nE);
}